// MPNN_57002805952758
// MI455X (gfx1250) — compile-verified
//
#include <hip/hip_runtime.h>

// MPNN forward for MI455X (gfx1250, wave32).
// fp32 end-to-end using V_WMMA_F32_16X16X4_F32; B operands staged in LDS
// (padded stride to avoid bank conflicts), scatter-add via native f32 atomics.
// LDS B addresses are made opaque per tile-iteration (asm token) so LLVM
// cannot hoist the loop-invariant ds_loads into registers and spill them
// to scratch (observed pathology in the previous round).

#define D 64
#define TD 129        // 2*D + 1
#define KPAD 132      // 33 chunks of K=4
#define BSTRIDE 65    // LDS row stride for edge-kernel B (64 + 1 pad)
#define GSTRIDE 193   // LDS row stride for GRU B (192 + 1 pad)

typedef __attribute__((ext_vector_type(2))) float v2f;
typedef __attribute__((ext_vector_type(8))) float v8f;

__device__ __forceinline__ v8f wmma4(v2f a, v2f b, v8f c) {
  // D = A(16x4 f32) * B(4x16 f32) + C(16x16 f32)
  return __builtin_amdgcn_wmma_f32_16x16x4_f32(
      /*neg_a=*/false, a, /*neg_b=*/false, b,
      /*c_mod=*/(short)0, c, /*reuse_a=*/false, /*reuse_b=*/false);
}

// One K=4 step for 4 adjacent 16-wide column tiles (edge kernel).
__device__ __forceinline__ void mac4(const float* __restrict__ brow, int row, v2f a,
                                     v8f& a0, v8f& a1, v8f& a2, v8f& a3) {
  v2f b0, b1, b2, b3;
  b0.x = brow[ 0 + row]; b0.y = brow[BSTRIDE +  0 + row];
  b1.x = brow[16 + row]; b1.y = brow[BSTRIDE + 16 + row];
  b2.x = brow[32 + row]; b2.y = brow[BSTRIDE + 32 + row];
  b3.x = brow[48 + row]; b3.y = brow[BSTRIDE + 48 + row];
  a0 = wmma4(a, b0, a0);
  a1 = wmma4(a, b1, a1);
  a2 = wmma4(a, b2, a2);
  a3 = wmma4(a, b3, a3);
}

__global__ void k_zero(float* __restrict__ p, int n) {
  int i = blockIdx.x * blockDim.x + threadIdx.x;
  int stride = gridDim.x * blockDim.x;
  for (; i < n; i += stride) p[i] = 0.0f;
}

// -------- Edge kernel: messages = relu([nf[src], nf[dst], ev] @ Wm^T + bm),
//          scatter-add into agg[dst]. One wave handles 16 edges per tile.
__global__ void __launch_bounds__(256)
k_edge(const float* __restrict__ nf, const float* __restrict__ ev,
       const int* __restrict__ src, const int* __restrict__ dst,
       const float* __restrict__ Wm, const float* __restrict__ bm,
       float* __restrict__ agg, int E) {
  __shared__ float ldsB[KPAD * BSTRIDE];  // ldsB[k][n] = Wm[n][k], rows >=129 are 0
  for (int i = threadIdx.x; i < KPAD * D; i += blockDim.x) {
    int k = i >> 6;
    int n = i & 63;
    ldsB[k * BSTRIDE + n] = (k < TD) ? Wm[n * TD + k] : 0.0f;
  }
  __syncthreads();

  const int lane  = threadIdx.x & 31;
  const int wave  = threadIdx.x >> 5;
  const int wpb   = blockDim.x >> 5;
  const int row   = lane & 15;      // A-matrix M row / B-matrix N col
  const int half  = lane >> 4;      // 0: K+0,1  1: K+2,3
  const int khalf = half * 2;
  const int mbase = half * 8;       // C/D rows for this lane-half
  const int nTiles = E >> 4;
  const int tstride = gridDim.x * wpb;

  for (int tile = blockIdx.x * wpb + wave; tile < nTiles; tile += tstride) {
    // Opaque zero: forces LDS B loads to stay inside this loop (no hoist/spill).
    int zofs = 0;
    asm volatile("" : "+v"(zofs));
    const float* ldsBo = ldsB + zofs;

    const int e0 = tile << 4;
    const int e  = e0 + row;
    const int s  = src[e];
    const int d  = dst[e];
    const float evv = ev[e];
    const float* srow = nf + (size_t)s * D;
    const float* drow = nf + (size_t)d * D;

    const float b0 = bm[ 0 + row], b1 = bm[16 + row];
    const float b2 = bm[32 + row], b3 = bm[48 + row];
    v8f acc0 = {b0,b0,b0,b0,b0,b0,b0,b0};
    v8f acc1 = {b1,b1,b1,b1,b1,b1,b1,b1};
    v8f acc2 = {b2,b2,b2,b2,b2,b2,b2,b2};
    v8f acc3 = {b3,b3,b3,b3,b3,b3,b3,b3};

    // Phase 1: K 0..63 (src features)
#pragma unroll 4
    for (int kc = 0; kc < 16; ++kc) {
      const int k = (kc << 2) + khalf;
      v2f a = *(const v2f*)(srow + k);
      mac4(ldsBo + k * BSTRIDE, row, a, acc0, acc1, acc2, acc3);
    }
    // Phase 2: K 64..127 (dst features)
#pragma unroll 4
    for (int kc = 16; kc < 32; ++kc) {
      const int k = (kc << 2) + khalf;
      v2f a = *(const v2f*)(drow + (k - 64));
      mac4(ldsBo + k * BSTRIDE, row, a, acc0, acc1, acc2, acc3);
    }
    // Phase 3: K 128 (edge value), 129..131 zero pad
    {
      const int k = 128 + khalf;
      v2f a;
      a.x = (half == 0) ? evv : 0.0f;
      a.y = 0.0f;
      mac4(ldsBo + k * BSTRIDE, row, a, acc0, acc1, acc2, acc3);
    }

    // ReLU + scatter-add: element (M = mbase+r, c = t*16+row)
#pragma unroll
    for (int r = 0; r < 8; ++r) {
      const int dnode = dst[e0 + mbase + r];
      float* dstrow = agg + (size_t)dnode * D;
      unsafeAtomicAdd(dstrow +  0 + row, fmaxf(acc0[r], 0.0f));
      unsafeAtomicAdd(dstrow + 16 + row, fmaxf(acc1[r], 0.0f));
      unsafeAtomicAdd(dstrow + 32 + row, fmaxf(acc2[r], 0.0f));
      unsafeAtomicAdd(dstrow + 48 + row, fmaxf(acc3[r], 0.0f));
    }
  }
}

// -------- GRU kernel: gi = agg@Wih^T+bih, gh = h@Whh^T+bhh, gates r/z/n,
//          updated summed directly into pooled[64]. One wave per 16-node tile.
__global__ void __launch_bounds__(256)
k_gru(const float* __restrict__ nf, const float* __restrict__ agg,
      const float* __restrict__ Wih, const float* __restrict__ Whh,
      const float* __restrict__ bih, const float* __restrict__ bhh,
      float* __restrict__ pooled, int N) {
  __shared__ float ldsBih[D * GSTRIDE];  // [k][j] = Wih[j][k], j in 0..191
  __shared__ float ldsBhh[D * GSTRIDE];
  for (int i = threadIdx.x; i < D * 192; i += blockDim.x) {
    int k = i / 192;
    int j = i - k * 192;
    ldsBih[k * GSTRIDE + j] = Wih[j * D + k];
    ldsBhh[k * GSTRIDE + j] = Whh[j * D + k];
  }
  __syncthreads();

  const int lane  = threadIdx.x & 31;
  const int wave  = threadIdx.x >> 5;
  const int wpb   = blockDim.x >> 5;
  const int row   = lane & 15;
  const int half  = lane >> 4;
  const int khalf = half * 2;
  const int mbase = half * 8;
  const int nTiles = N >> 4;
  const int tstride = gridDim.x * wpb;

  float partial[4] = {0.0f, 0.0f, 0.0f, 0.0f};

  for (int tile = blockIdx.x * wpb + wave; tile < nTiles; tile += tstride) {
    // Opaque zero: keep LDS B loads inside the tile loop (no hoist/spill).
    int zofs = 0;
    asm volatile("" : "+v"(zofs));
    const float* Bih = ldsBih + zofs;
    const float* Bhh = ldsBhh + zofs;

    const int n0 = tile << 4;
    const float* arow = agg + (size_t)(n0 + row) * D;
    const float* hrow = nf  + (size_t)(n0 + row) * D;

#pragma unroll
    for (int t = 0; t < 4; ++t) {
      const int cb = t * 16 + row;  // output channel within D
      const float vir = bih[  0 + cb], viz = bih[ 64 + cb], vin = bih[128 + cb];
      const float vhr = bhh[  0 + cb], vhz = bhh[ 64 + cb], vhn = bhh[128 + cb];
      v8f ir  = {vir,vir,vir,vir,vir,vir,vir,vir};
      v8f iz  = {viz,viz,viz,viz,viz,viz,viz,viz};
      v8f in_ = {vin,vin,vin,vin,vin,vin,vin,vin};
      v8f hr  = {vhr,vhr,vhr,vhr,vhr,vhr,vhr,vhr};
      v8f hz  = {vhz,vhz,vhz,vhz,vhz,vhz,vhz,vhz};
      v8f hn  = {vhn,vhn,vhn,vhn,vhn,vhn,vhn,vhn};

#pragma unroll 4
      for (int kc = 0; kc < 16; ++kc) {
        const int k = (kc << 2) + khalf;
        v2f aa = *(const v2f*)(arow + k);
        v2f ah = *(const v2f*)(hrow + k);
        const float* bi = Bih + k * GSTRIDE;
        const float* bh = Bhh + k * GSTRIDE;
        v2f b_ir = {bi[  0 + cb], bi[GSTRIDE +   0 + cb]};
        v2f b_iz = {bi[ 64 + cb], bi[GSTRIDE +  64 + cb]};
        v2f b_in = {bi[128 + cb], bi[GSTRIDE + 128 + cb]};
        v2f b_hr = {bh[  0 + cb], bh[GSTRIDE +   0 + cb]};
        v2f b_hz = {bh[ 64 + cb], bh[GSTRIDE +  64 + cb]};
        v2f b_hn = {bh[128 + cb], bh[GSTRIDE + 128 + cb]};
        ir  = wmma4(aa, b_ir, ir);
        iz  = wmma4(aa, b_iz, iz);
        in_ = wmma4(aa, b_in, in_);
        hr  = wmma4(ah, b_hr, hr);
        hz  = wmma4(ah, b_hz, hz);
        hn  = wmma4(ah, b_hn, hn);
      }

      float psum = 0.0f;
#pragma unroll
      for (int r = 0; r < 8; ++r) {
        const int m = mbase + r;
        const float hval = nf[(size_t)(n0 + m) * D + cb];
        const float rg = 1.0f / (1.0f + __expf(-(ir[r] + hr[r])));
        const float zg = 1.0f / (1.0f + __expf(-(iz[r] + hz[r])));
        const float ng = tanhf(in_[r] + rg * hn[r]);
        psum += (1.0f - zg) * ng + zg * hval;
      }
      partial[t] += psum;
    }
  }

#pragma unroll
  for (int t = 0; t < 4; ++t)
    unsafeAtomicAdd(pooled + t * 16 + row, partial[t]);
}

// -------- Readout: q = Wpol@pooled + bpol (64), v = Wval@pooled + bval (1)
__global__ void k_readout(const float* __restrict__ pooled,
                          const float* __restrict__ Wpol, const float* __restrict__ bpol,
                          const float* __restrict__ Wval, const float* __restrict__ bval,
                          float* __restrict__ out) {
  __shared__ float p[D];
  if (threadIdx.x < D) p[threadIdx.x] = pooled[threadIdx.x];
  __syncthreads();
  const int i = threadIdx.x;
  if (i < 64) {
    float s = bpol[i];
#pragma unroll 8
    for (int k = 0; k < D; ++k) s += Wpol[i * D + k] * p[k];
    out[i] = s;
  } else if (i == 64) {
    float s = bval[0];
#pragma unroll 8
    for (int k = 0; k < D; ++k) s += Wval[k] * p[k];
    out[64] = s;
  }
}

extern "C" void kernel_launch(void* const* d_in, const int* in_sizes, int n_in,
                              void* d_out, int out_size, void* d_ws, size_t ws_size,
                              hipStream_t stream) {
  const float* nf   = (const float*)d_in[0];
  const float* ev   = (const float*)d_in[1];
  const int*   src  = (const int*)d_in[2];
  const int*   dst  = (const int*)d_in[3];
  const float* Wm   = (const float*)d_in[4];
  const float* bm   = (const float*)d_in[5];
  const float* Wih  = (const float*)d_in[6];
  const float* Whh  = (const float*)d_in[7];
  const float* bih  = (const float*)d_in[8];
  const float* bhh  = (const float*)d_in[9];
  const float* Wpol = (const float*)d_in[10];
  const float* bpol = (const float*)d_in[11];
  const float* Wval = (const float*)d_in[12];
  const float* bval = (const float*)d_in[13];

  const int N = in_sizes[0] / D;
  const int E = in_sizes[1];

  float* agg    = (float*)d_ws;            // [N, D]
  float* pooled = agg + (size_t)N * D;     // [D]

  k_zero   <<<2048, 256, 0, stream>>>(agg, N * D + D);
  k_edge   <<<768, 256, 0, stream>>>(nf, ev, src, dst, Wm, bm, agg, E);
  k_gru    <<<256, 256, 0, stream>>>(nf, agg, Wih, Whh, bih, bhh, pooled, N);
  k_readout<<<1, 128, 0, stream>>>(pooled, Wpol, bpol, Wval, bval, (float*)d_out);
}